// GuidedBlur_84567906058644
// MI455X (gfx1250) — compile-verified
//
#include <hip/hip_runtime.h>
#include <cstddef>

// Problem constants (from reference)
constexpr int   NB   = 8;     // batch
constexpr int   CC   = 3;     // guidance channels == input channels
constexpr int   HH   = 512;
constexpr int   WW   = 512;
constexpr int   RR   = 2;     // kernel radius (K=5)
constexpr float EPSV = 1e-4f;

// Tiling
constexpr int OT   = 32;             // output tile (32x32)
constexpr int EXTT = OT + 2 * RR;    // 36: region where a,b are needed
constexpr int INT_ = OT + 4 * RR;    // 40: guidance/input tile incl. halo
constexpr int PS   = 6;              // floats per tile pixel (g0,g1,g2,p0,p1,p2)
constexpr int ABST = 12;             // floats per a/b record (a[3][3], b[3])
constexpr int NTH  = 256;            // 8 wave32s

// LDS layout (floats):
//   tile : [40*40][6]        = 9600   (byte off 0,      38400 B)
//   ab   : [36*36][12]       = 15552  (byte off 38400,  62208 B)  16B-aligned records
//   habs : [36*32][12]       = 13824  (byte off 100608, 55296 B)  16B-aligned records
constexpr int TILE_FLOATS = INT_ * INT_ * PS;     // 9600
constexpr int AB_FLOATS   = EXTT * EXTT * ABST;   // 15552
constexpr int HABS_FLOATS = EXTT * OT * ABST;     // 13824
constexpr size_t SMEM_BYTES =
    (size_t)(TILE_FLOATS + AB_FLOATS + HABS_FLOATS) * 4;   // 155904 (2 WGs/WGP)

__device__ __forceinline__ int refl(int i, int n) {
    i = (i < 0) ? -i : i;                 // reflect about 0 (no edge repeat)
    i = (i >= n) ? (2 * n - 2 - i) : i;   // reflect about n-1
    return i;
}

// CDNA5 async global -> LDS copy (per-lane gather), tracked by ASYNCcnt.
__device__ __forceinline__ void async_g2l_b32(const float* src, unsigned lds_byte_off) {
#if defined(__gfx1250__)
    asm volatile("global_load_async_to_lds_b32 %0, %1, off"
                 :: "v"(lds_byte_off), "v"(src)
                 : "memory");
#else
    (void)src; (void)lds_byte_off;
#endif
}

__device__ __forceinline__ void wait_async0() {
#if defined(__gfx1250__)
    asm volatile("s_wait_asynccnt 0x0" ::: "memory");
#endif
}

// Given 5x5 window sums m[21] (sg0..2, sp0..2, s00,s01,s02,s11,s12,s22,
// q00..q22 row-major g_c*p_p), solve the regularized 3x3 SPD system and
// store a[3][3] row-major + b[3] to the 12-float LDS record q.
__device__ __forceinline__ void solve_and_store(const float m[21], float* q) {
    const float n = 1.0f / 25.0f;
    const float mI0 = m[0]*n, mI1 = m[1]*n, mI2 = m[2]*n;
    const float mp0 = m[3]*n, mp1 = m[4]*n, mp2 = m[5]*n;

    const float v00 = fmaf(m[ 6], n, -mI0*mI0) + EPSV;
    const float v01 = fmaf(m[ 7], n, -mI0*mI1);
    const float v02 = fmaf(m[ 8], n, -mI0*mI2);
    const float v11 = fmaf(m[ 9], n, -mI1*mI1) + EPSV;
    const float v12 = fmaf(m[10], n, -mI1*mI2);
    const float v22 = fmaf(m[11], n, -mI2*mI2) + EPSV;

    const float c00 = fmaf(m[12], n, -mI0*mp0);
    const float c01 = fmaf(m[13], n, -mI0*mp1);
    const float c02 = fmaf(m[14], n, -mI0*mp2);
    const float c10 = fmaf(m[15], n, -mI1*mp0);
    const float c11 = fmaf(m[16], n, -mI1*mp1);
    const float c12 = fmaf(m[17], n, -mI1*mp2);
    const float c20 = fmaf(m[18], n, -mI2*mp0);
    const float c21 = fmaf(m[19], n, -mI2*mp1);
    const float c22 = fmaf(m[20], n, -mI2*mp2);

    // inverse of symmetric 3x3 via adjugate
    const float i00 = v11*v22 - v12*v12;
    const float i01 = v02*v12 - v01*v22;
    const float i02 = v01*v12 - v02*v11;
    const float i11 = v00*v22 - v02*v02;
    const float i12 = v01*v02 - v00*v12;
    const float i22 = v00*v11 - v01*v01;
    const float det = v00*i00 + v01*i01 + v02*i02;
    const float rd  = 1.0f / det;

    const float a00 = (i00*c00 + i01*c10 + i02*c20) * rd;
    const float a01 = (i00*c01 + i01*c11 + i02*c21) * rd;
    const float a02 = (i00*c02 + i01*c12 + i02*c22) * rd;
    const float a10 = (i01*c00 + i11*c10 + i12*c20) * rd;
    const float a11 = (i01*c01 + i11*c11 + i12*c21) * rd;
    const float a12 = (i01*c02 + i11*c12 + i12*c22) * rd;
    const float a20 = (i02*c00 + i12*c10 + i22*c20) * rd;
    const float a21 = (i02*c01 + i12*c11 + i22*c21) * rd;
    const float a22 = (i02*c02 + i12*c12 + i22*c22) * rd;

    q[0]=a00; q[1]=a01; q[2]=a02;
    q[3]=a10; q[4]=a11; q[5]=a12;
    q[6]=a20; q[7]=a21; q[8]=a22;
    q[ 9] = mp0 - (a00*mI0 + a10*mI1 + a20*mI2);
    q[10] = mp1 - (a01*mI0 + a11*mI1 + a21*mI2);
    q[11] = mp2 - (a02*mI0 + a12*mI1 + a22*mI2);
}

extern "C" __global__ __launch_bounds__(NTH)
void guided_filter_fused(const float* __restrict__ gsrc,
                         const float* __restrict__ psrc,
                         float* __restrict__ out)
{
    extern __shared__ float smem[];
    float* tile = smem;                              // [INT_*INT_][PS]
    float* ab   = smem + TILE_FLOATS;                // [EXTT*EXTT][ABST]
    float* habs = smem + TILE_FLOATS + AB_FLOATS;    // [EXTT*OT][ABST]

    const int tid = threadIdx.x;
    const int tx0 = blockIdx.x * OT;
    const int ty0 = blockIdx.y * OT;
    const int bz  = blockIdx.z;
    const int HWp = HH * WW;

    const float* gb = gsrc + (size_t)bz * CC * HWp;
    const float* pb = psrc + (size_t)bz * CC * HWp;

    // ---------------- Phase 0: reflected tile load via async global->LDS ----
    for (int i = tid; i < INT_ * INT_; i += NTH) {
        const int ly  = i / INT_;
        const int lx  = i - ly * INT_;
        const int gy  = refl(ty0 - 2 * RR + ly, HH);
        const int gx  = refl(tx0 - 2 * RR + lx, WW);
        const int pix = gy * WW + gx;
        const unsigned dst = (unsigned)(i * PS) * 4u;
        const float* s0 = gb + pix;
        const float* s1 = pb + pix;
        async_g2l_b32(s0,            dst +  0);
        async_g2l_b32(s0 +     HWp,  dst +  4);
        async_g2l_b32(s0 + 2 * HWp,  dst +  8);
        async_g2l_b32(s1,            dst + 12);
        async_g2l_b32(s1 +     HWp,  dst + 16);
        async_g2l_b32(s1 + 2 * HWp,  dst + 20);
    }
    wait_async0();       // my wave's async copies are in LDS
    __syncthreads();     // ...and visible to all waves

    // ---------------- Phase A: moments + solve, two row-adjacent pixels/thread
    // Row-sums over the 5-column window are computed once per tile row and
    // shared by the two vertically adjacent 5x5 windows (rows t=0..4 / 1..5).
    for (int u = tid; u < (EXTT / 2) * EXTT; u += NTH) {   // 18*36 = 648 pairs
        const int pr = u / EXTT;
        const int ex = u - pr * EXTT;
        const int ey = pr * 2;

        float acc0[21], acc1[21], rs[21];
        #pragma unroll
        for (int j = 0; j < 21; ++j) { acc0[j] = 0.f; acc1[j] = 0.f; }

        #pragma unroll
        for (int t = 0; t < 6; ++t) {          // tile rows ey+t
            #pragma unroll
            for (int j = 0; j < 21; ++j) rs[j] = 0.f;
            #pragma unroll
            for (int dx = 0; dx < 5; ++dx) {
                const float* tp = tile + ((ey + t) * INT_ + (ex + dx)) * PS;
                const float2 v0 = *reinterpret_cast<const float2*>(tp);      // g0 g1
                const float2 v1 = *reinterpret_cast<const float2*>(tp + 2);  // g2 p0
                const float2 v2 = *reinterpret_cast<const float2*>(tp + 4);  // p1 p2
                const float g0 = v0.x, g1 = v0.y, g2 = v1.x;
                const float p0 = v1.y, p1 = v2.x, p2 = v2.y;
                rs[0] += g0; rs[1] += g1; rs[2] += g2;
                rs[3] += p0; rs[4] += p1; rs[5] += p2;
                rs[ 6] = fmaf(g0,g0,rs[ 6]); rs[ 7] = fmaf(g0,g1,rs[ 7]);
                rs[ 8] = fmaf(g0,g2,rs[ 8]); rs[ 9] = fmaf(g1,g1,rs[ 9]);
                rs[10] = fmaf(g1,g2,rs[10]); rs[11] = fmaf(g2,g2,rs[11]);
                rs[12] = fmaf(g0,p0,rs[12]); rs[13] = fmaf(g0,p1,rs[13]);
                rs[14] = fmaf(g0,p2,rs[14]); rs[15] = fmaf(g1,p0,rs[15]);
                rs[16] = fmaf(g1,p1,rs[16]); rs[17] = fmaf(g1,p2,rs[17]);
                rs[18] = fmaf(g2,p0,rs[18]); rs[19] = fmaf(g2,p1,rs[19]);
                rs[20] = fmaf(g2,p2,rs[20]);
            }
            if (t < 5) {
                #pragma unroll
                for (int j = 0; j < 21; ++j) acc0[j] += rs[j];
            }
            if (t > 0) {
                #pragma unroll
                for (int j = 0; j < 21; ++j) acc1[j] += rs[j];
            }
        }

        solve_and_store(acc0, ab + ( ey      * EXTT + ex) * ABST);
        solve_and_store(acc1, ab + ((ey + 1) * EXTT + ex) * ABST);
    }
    __syncthreads();

    // ---------------- Phase B1: horizontal 5-sums of a,b (separable blur) ----
    for (int u = tid; u < EXTT * OT; u += NTH) {           // 36*32 = 1152
        const int ey = u / OT;
        const int ox = u - ey * OT;

        float acc[12];
        #pragma unroll
        for (int j = 0; j < 12; ++j) acc[j] = 0.f;

        #pragma unroll
        for (int dx = 0; dx < 5; ++dx) {
            const float* q = ab + (ey * EXTT + (ox + dx)) * ABST;
            const float4 q0 = *reinterpret_cast<const float4*>(q);
            const float4 q1 = *reinterpret_cast<const float4*>(q + 4);
            const float4 q2 = *reinterpret_cast<const float4*>(q + 8);
            acc[0]+=q0.x; acc[1]+=q0.y; acc[ 2]+=q0.z; acc[ 3]+=q0.w;
            acc[4]+=q1.x; acc[5]+=q1.y; acc[ 6]+=q1.z; acc[ 7]+=q1.w;
            acc[8]+=q2.x; acc[9]+=q2.y; acc[10]+=q2.z; acc[11]+=q2.w;
        }

        float* h = habs + (ey * OT + ox) * ABST;
        float4 w0, w1, w2;
        w0.x=acc[0]; w0.y=acc[1]; w0.z=acc[ 2]; w0.w=acc[ 3];
        w1.x=acc[4]; w1.y=acc[5]; w1.z=acc[ 6]; w1.w=acc[ 7];
        w2.x=acc[8]; w2.y=acc[9]; w2.z=acc[10]; w2.w=acc[11];
        *reinterpret_cast<float4*>(h    ) = w0;
        *reinterpret_cast<float4*>(h + 4) = w1;
        *reinterpret_cast<float4*>(h + 8) = w2;
    }
    __syncthreads();

    // ---------------- Phase B2: vertical 5-sums + combine with guidance -----
    const float inv25 = 1.0f / 25.0f;
    for (int o = tid; o < OT * OT; o += NTH) {
        const int oy = o / OT;
        const int ox = o - oy * OT;

        float acc[12];
        #pragma unroll
        for (int j = 0; j < 12; ++j) acc[j] = 0.f;

        #pragma unroll
        for (int dy = 0; dy < 5; ++dy) {
            const float* h = habs + ((oy + dy) * OT + ox) * ABST;
            const float4 q0 = *reinterpret_cast<const float4*>(h);
            const float4 q1 = *reinterpret_cast<const float4*>(h + 4);
            const float4 q2 = *reinterpret_cast<const float4*>(h + 8);
            acc[0]+=q0.x; acc[1]+=q0.y; acc[ 2]+=q0.z; acc[ 3]+=q0.w;
            acc[4]+=q1.x; acc[5]+=q1.y; acc[ 6]+=q1.z; acc[ 7]+=q1.w;
            acc[8]+=q2.x; acc[9]+=q2.y; acc[10]+=q2.z; acc[11]+=q2.w;
        }

        // center guidance: tile local (oy+4, ox+4)
        const float* t = tile + ((oy + 2*RR) * INT_ + (ox + 2*RR)) * PS;
        const float g0 = t[0], g1 = t[1], g2 = t[2];

        const int gy = ty0 + oy, gx = tx0 + ox;
        const size_t ob = ((size_t)bz * CC * HH + (size_t)gy) * WW + gx;
        out[ob          ] = (g0*acc[0] + g1*acc[3] + g2*acc[6] + acc[ 9]) * inv25;
        out[ob +     HWp] = (g0*acc[1] + g1*acc[4] + g2*acc[7] + acc[10]) * inv25;
        out[ob + 2 * HWp] = (g0*acc[2] + g1*acc[5] + g2*acc[8] + acc[11]) * inv25;
    }
}

extern "C" void kernel_launch(void* const* d_in, const int* in_sizes, int n_in,
                              void* d_out, int out_size, void* d_ws, size_t ws_size,
                              hipStream_t stream) {
    (void)in_sizes; (void)n_in; (void)out_size; (void)d_ws; (void)ws_size;
    const float* guidance = (const float*)d_in[0];
    const float* input    = (const float*)d_in[1];
    float* out            = (float*)d_out;

    dim3 grid(WW / OT, HH / OT, NB);   // 16 x 16 x 8
    dim3 block(NTH);
    guided_filter_fused<<<grid, block, SMEM_BYTES, stream>>>(guidance, input, out);
}